// Multi_head_SelfAttentionPredictModel_parallel_68607807586678
// MI455X (gfx1250) — compile-verified
//
#include <hip/hip_runtime.h>
#include <hip/hip_bf16.h>
#include <math.h>

// ---------------------------------------------------------------------------
// Problem dimensions (fixed by the reference)
// ---------------------------------------------------------------------------
constexpr int Bz = 2;          // batch
constexpr int Sq = 2048;       // sequence length
constexpr int Hd = 2048;       // hidden
constexpr int NH = 16;         // heads
constexpr int HD = 128;        // head dim
constexpr float NEGV = -1000000.0f;

typedef float v2f __attribute__((ext_vector_type(2)));
typedef float v8f __attribute__((ext_vector_type(8)));

// D = A(16x4 f32) * B(4x16 f32) + C(16x16 f32)  -- full fp32 matrix pipe
#define WMMA_F32(a, b, c) \
  __builtin_amdgcn_wmma_f32_16x16x4_f32(false, (a), false, (b), (short)0, (c), false, false)

__device__ __forceinline__ v8f vzero8() {
  v8f z = {0.f, 0.f, 0.f, 0.f, 0.f, 0.f, 0.f, 0.f};
  return z;
}

// ---------------------------------------------------------------------------
// GEMM: C[M,N] = A[M,K] * B[K,N], all fp32, M,N,K multiples of tile sizes.
// Workgroup = 256 threads (8 waves). Tile 128(M) x 128(N), K staged by 32.
// Wave grid 4(M) x 2(N): each wave computes 32x64 via 2x4 WMMA accumulators.
//
// B tile is staged k-pair interleaved: element (k, col) -> Bsp[k/2][2*col+(k&1)]
// so a B fragment {(k,col),(k+1,col)} is one contiguous ds_load_b64.
// Row stride 2*TN+32 = 288 floats == 32 banks (mod 64): lanes 16-31 (k+2,k+3)
// land in the disjoint upper bank half -> conflict-free b64 reads.
// ---------------------------------------------------------------------------
constexpr int TM = 128, TN = 128, TK = 32;
constexpr int APAD = 4;     // A stride 36 floats: 36*lm mod 64 distinct for lm=0..15
constexpr int BIPAD = 32;   // B interleaved k-half-row stride 2*TN+32 = 288 floats

__global__ __launch_bounds__(256) void gemm_f32_wmma(
    const float* __restrict__ A, const float* __restrict__ Bm,
    float* __restrict__ C, int M, int N, int K) {
  __shared__ float As[TM][TK + APAD];
  __shared__ float Bsp[TK / 2][2 * TN + BIPAD];

  const int t = threadIdx.x;
  const int lane = t & 31;
  const int wave = t >> 5;
  const int lm = lane & 15;      // row/col within 16-wide fragment
  const int lh = lane >> 4;      // lane-half selects K pair (ISA A/B f32 layout)
  const int wm = wave & 3;       // 32-row band within 128
  const int wn = wave >> 2;      // 64-col band within 128
  const int m0 = blockIdx.y * TM;
  const int n0 = blockIdx.x * TN;

  v8f acc[2][4];
#pragma unroll
  for (int i = 0; i < 2; ++i)
#pragma unroll
    for (int j = 0; j < 4; ++j) acc[i][j] = vzero8();

  for (int kt = 0; kt < K; kt += TK) {
    // Stage A tile: 128x32 floats = 1024 float4, 4 per thread (b128 LDS stores)
#pragma unroll
    for (int i = 0; i < 4; ++i) {
      int idx = t + i * 256;
      int r = idx >> 3;
      int c = (idx & 7) << 2;
      float4 v = *(const float4*)(A + (size_t)(m0 + r) * K + kt + c);
      *(float4*)&As[r][c] = v;
    }
    // Stage B tile: 32x128 floats, scattered into k-pair interleaved layout
#pragma unroll
    for (int i = 0; i < 4; ++i) {
      int idx = t + i * 256;
      int r = idx >> 5;
      int c = (idx & 31) << 2;
      float4 v = *(const float4*)(Bm + (size_t)(kt + r) * N + n0 + c);
      float* dst = &Bsp[r >> 1][2 * c + (r & 1)];
      dst[0] = v.x;
      dst[2] = v.y;
      dst[4] = v.z;
      dst[6] = v.w;
    }
    __syncthreads();

#pragma unroll
    for (int kk = 0; kk < TK; kk += 4) {
      // A fragment (16x4): lanes 0-15 hold k=kk,kk+1 ; lanes 16-31 k=kk+2,kk+3
      v2f af[2];
#pragma unroll
      for (int im = 0; im < 2; ++im) {
        int row = wm * 32 + im * 16 + lm;
        af[im] = *(const v2f*)&As[row][kk + 2 * lh];
      }
      const int khalf = (kk >> 1) + lh;
#pragma unroll
      for (int j = 0; j < 4; ++j) {
        int col = wn * 64 + j * 16 + lm;
        v2f bf = *(const v2f*)&Bsp[khalf][2 * col];
        acc[0][j] = WMMA_F32(af[0], bf, acc[0][j]);
        acc[1][j] = WMMA_F32(af[1], bf, acc[1][j]);
      }
    }
    __syncthreads();
  }

  // C/D layout: vgpr r -> row (r + 8*lane_half), lane%16 -> col
#pragma unroll
  for (int im = 0; im < 2; ++im)
#pragma unroll
    for (int j = 0; j < 4; ++j)
#pragma unroll
      for (int r = 0; r < 8; ++r) {
        int m = m0 + wm * 32 + im * 16 + r + 8 * lh;
        int n = n0 + wn * 64 + j * 16 + lm;
        C[(size_t)m * N + n] = acc[im][j][r];
      }
}

// ---------------------------------------------------------------------------
// Flash attention, fp32 WMMA. One workgroup (4 waves / 128 thr) owns one
// (b, h) and 64 q rows (16 per wave). K/V staged in LDS, 32 keys per tile.
// Reference reshapes without transpose: head h, seq s lives at flat row h*S+s.
// K tile: row-major (score B-frags read K^T columns as natural b64).
// V tile: k-pair interleaved (PV B-frags as single b64), like the GEMM B tile.
// ---------------------------------------------------------------------------
constexpr int QT = 64;     // q rows per workgroup
constexpr int KT = 32;     // keys per tile
constexpr int KVP = 12;    // K LDS pad (stride 140 -> distinct banks per lane)
constexpr int VIPAD = 32;  // V interleaved k-half-row stride 2*HD+32 = 288 floats
constexpr int PPAD = 4;    // P LDS pad (stride 36)

__global__ __launch_bounds__(128) void attn_f32_wmma(
    const float* __restrict__ Qm, const float* __restrict__ Km,
    const float* __restrict__ Vm, const int* __restrict__ mask,
    float* __restrict__ O) {
  __shared__ float Ks[KT][HD + KVP];
  __shared__ float Vsp[KT / 2][2 * HD + VIPAD];
  __shared__ float Ps[4][16][KT + PPAD];
  __shared__ float corrS[4][16];
  __shared__ float rsumS[4][16];
  __shared__ int mkS[KT];

  const int t = threadIdx.x;
  const int lane = t & 31;
  const int w = t >> 5;
  const int lm = lane & 15;
  const int lh = lane >> 4;

  int bid = blockIdx.x;
  const int nqb = Sq / QT;
  const int qb = bid % nqb; bid /= nqb;
  const int h = bid % NH;
  const int b = bid / NH;
  const int q0 = qb * QT;

  const size_t baseBH = (size_t)b * Sq * Hd + (size_t)h * Sq * HD;

  // Q fragments for this wave's 16 rows, kept in registers (32 d-steps of 4)
  const float* qrow = Qm + baseBH + (size_t)(q0 + w * 16 + lm) * HD;
  v2f qf[HD / 4];
#pragma unroll
  for (int kd = 0; kd < HD / 4; ++kd)
    qf[kd] = *(const v2f*)(qrow + 4 * kd + 2 * lh);

  // per-lane row masks for accumulator rows m = r + 8*lh
  int mq[8];
#pragma unroll
  for (int r = 0; r < 8; ++r)
    mq[r] = mask[b * Sq + q0 + w * 16 + r + 8 * lh];

  v8f o[8];
#pragma unroll
  for (int s = 0; s < 8; ++s) o[s] = vzero8();
  float rmax = -INFINITY;
  float rsum = 0.0f;

  for (int kt = 0; kt < Sq; kt += KT) {
    // Stage K (row-major) and V (k-pair interleaved): 32x128 floats each
#pragma unroll
    for (int i = 0; i < 8; ++i) {
      int idx = t + i * 128;
      int r = idx >> 5;
      int c = (idx & 31) << 2;
      float4 kv = *(const float4*)(Km + baseBH + (size_t)(kt + r) * HD + c);
      *(float4*)&Ks[r][c] = kv;
      float4 vv = *(const float4*)(Vm + baseBH + (size_t)(kt + r) * HD + c);
      float* dst = &Vsp[r >> 1][2 * c + (r & 1)];
      dst[0] = vv.x;
      dst[2] = vv.y;
      dst[4] = vv.z;
      dst[6] = vv.w;
    }
    if (t < KT) mkS[t] = mask[b * Sq + kt + t];
    __syncthreads();

    // scores S(16q x 32k) = Q * K^T : contraction over d, B[d][k] = K[k][d]
    v8f sc[2];
    sc[0] = vzero8();
    sc[1] = vzero8();
#pragma unroll
    for (int kd = 0; kd < HD / 4; ++kd) {
      v2f af = qf[kd];
#pragma unroll
      for (int j = 0; j < 2; ++j) {
        int kc = j * 16 + lm;
        v2f bf = *(const v2f*)&Ks[kc][4 * kd + 2 * lh];
        sc[j] = WMMA_F32(af, bf, sc[j]);
      }
    }

    // apply mask, spill raw scores to LDS (per-wave scratch)
#pragma unroll
    for (int j = 0; j < 2; ++j)
#pragma unroll
      for (int r = 0; r < 8; ++r) {
        int m = r + 8 * lh;
        int kc = j * 16 + lm;
        float v = sc[j][r];
        if (mq[r] == 0 || mkS[kc] == 0) v = NEGV;
        Ps[w][m][kc] = v;
      }
    __syncthreads();

    // online softmax: lane owns row lm, half the columns; combine via shfl
    {
      const int c0 = lh * 16;
      float pm = -INFINITY;
#pragma unroll
      for (int c = 0; c < 16; ++c) pm = fmaxf(pm, Ps[w][lm][c0 + c]);
      pm = fmaxf(pm, __shfl_xor(pm, 16, 32));
      float nm = fmaxf(rmax, pm);
      float cf = __expf(rmax - nm);   // exp(-inf)=0 on first tile
      float ps = 0.0f;
#pragma unroll
      for (int c = 0; c < 16; ++c) {
        float e = __expf(Ps[w][lm][c0 + c] - nm);
        Ps[w][lm][c0 + c] = e;
        ps += e;
      }
      ps += __shfl_xor(ps, 16, 32);
      rsum = rsum * cf + ps;
      rmax = nm;
      if (lane < 16) corrS[w][lm] = cf;
    }
    __syncthreads();

    // rescale running output accumulators
    float cv[8];
#pragma unroll
    for (int r = 0; r < 8; ++r) cv[r] = corrS[w][r + 8 * lh];
#pragma unroll
    for (int s = 0; s < 8; ++s)
#pragma unroll
      for (int r = 0; r < 8; ++r) o[s][r] *= cv[r];

    // O(16q x 128d) += P(16q x 32k) * V(32k x 128d)
#pragma unroll
    for (int kk = 0; kk < KT; kk += 4) {
      v2f af = *(const v2f*)&Ps[w][lm][kk + 2 * lh];
      const int khalf = (kk >> 1) + lh;
#pragma unroll
      for (int s = 0; s < 8; ++s) {
        int col = s * 16 + lm;
        v2f bf = *(const v2f*)&Vsp[khalf][2 * col];
        o[s] = WMMA_F32(af, bf, o[s]);
      }
    }
    __syncthreads();
  }

  if (lane < 16) rsumS[w][lm] = rsum;
  __syncthreads();
  float rv[8];
#pragma unroll
  for (int r = 0; r < 8; ++r) rv[r] = 1.0f / rsumS[w][r + 8 * lh];

#pragma unroll
  for (int s = 0; s < 8; ++s)
#pragma unroll
    for (int r = 0; r < 8; ++r) {
      int s2 = q0 + w * 16 + r + 8 * lh;
      int d = s * 16 + lm;
      O[baseBH + (size_t)s2 * HD + d] = o[s][r] * rv[r];
    }
}

// ---------------------------------------------------------------------------
// Host orchestration: 7 GEMMs + attention, ping-ponging 4 workspace slots.
// ---------------------------------------------------------------------------
extern "C" void kernel_launch(void* const* d_in, const int* in_sizes, int n_in,
                              void* d_out, int out_size, void* d_ws, size_t ws_size,
                              hipStream_t stream) {
  const float* X    = (const float*)d_in[0];
  const int*   mask = (const int*)d_in[1];
  const float* Wq   = (const float*)d_in[2];
  const float* Wk   = (const float*)d_in[3];
  const float* Wv   = (const float*)d_in[4];
  const float* Wqm  = (const float*)d_in[5];
  const float* Wkm  = (const float*)d_in[6];
  const float* Wvm  = (const float*)d_in[7];
  const float* Wo   = (const float*)d_in[8];
  float* out = (float*)d_out;

  const size_t SZ = (size_t)Bz * Sq * Hd;  // elements per activation buffer
  float* s0 = (float*)d_ws;
  float* s1 = s0 + SZ;
  float* s2 = s1 + SZ;
  float* s3 = s2 + SZ;

  const int M = Bz * Sq, N = Hd, K = Hd;
  dim3 gg(N / TN, M / TM);
  dim3 gb(256);

  gemm_f32_wmma<<<gg, gb, 0, stream>>>(X,  Wq,  s0, M, N, K);  // Q
  gemm_f32_wmma<<<gg, gb, 0, stream>>>(X,  Wk,  s1, M, N, K);  // K
  gemm_f32_wmma<<<gg, gb, 0, stream>>>(X,  Wv,  s2, M, N, K);  // V
  gemm_f32_wmma<<<gg, gb, 0, stream>>>(s0, Wqm, s3, M, N, K);  // Q_m
  gemm_f32_wmma<<<gg, gb, 0, stream>>>(s1, Wkm, s0, M, N, K);  // K_m (Q dead)
  gemm_f32_wmma<<<gg, gb, 0, stream>>>(s2, Wvm, s1, M, N, K);  // V_m (K dead)

  attn_f32_wmma<<<Bz * NH * (Sq / QT), 128, 0, stream>>>(s3, s0, s1, mask, s2);

  gemm_f32_wmma<<<gg, gb, 0, stream>>>(s2, Wo, out, M, N, K);  // final projection
}